// GraphSAGE_33148557591078
// MI455X (gfx1250) — compile-verified
//
#include <hip/hip_runtime.h>
#include <hip/hip_bf16.h>

// GraphSAGE (2-layer, mean aggregation) for MI455X / gfx1250.
//  - edge-parallel gather + L2-resident f32 atomics for aggregation
//    (x/h/sum/edges all fit in the 192MB L2)
//  - fp32 WMMA (V_WMMA_F32_16X16X4_F32) for fused  mean@W_l + x@W_r + b
//  - weights pre-transposed once, then staged global->LDS with the
//    Tensor Data Mover (tensor_load_to_lds + s_wait_tensorcnt)

#define NNODES 100000
#define NEDGES 1600000
#define DIN    128
#define DHID   128
#define DOUT   64

typedef __attribute__((ext_vector_type(2))) float v2f;
typedef __attribute__((ext_vector_type(8))) float v8f;
typedef __attribute__((ext_vector_type(4))) unsigned int u32x4;
typedef __attribute__((ext_vector_type(8))) int i32x8;
typedef __attribute__((ext_vector_type(4))) int i32x4;

// ---------------------------------------------------------------- utilities
__global__ void zero_kernel(float* __restrict__ p, long long n) {
  long long i = (long long)blockIdx.x * blockDim.x + threadIdx.x;
  if (i < n) p[i] = 0.0f;
}

// degree[dst] += 1 per edge (graph identical for both layers -> compute once)
__global__ void degree_kernel(const long long* __restrict__ ei,
                              float* __restrict__ deg, int E) {
  int e = blockIdx.x * blockDim.x + threadIdx.x;
  if (e < E) {
    int dst = (int)ei[(long long)E + e];
    (void)__hip_atomic_fetch_add(&deg[dst], 1.0f,
                                 __ATOMIC_RELAXED, __HIP_MEMORY_SCOPE_AGENT);
  }
}

// One wave per edge: 32 lanes x float4 = 128 features.
__global__ void scatter_kernel(const float* __restrict__ x,
                               const long long* __restrict__ ei,
                               float* __restrict__ sum, int E) {
  int e    = blockIdx.x * (blockDim.x >> 5) + (threadIdx.x >> 5);
  int lane = threadIdx.x & 31;
  if (e < E) {
    int src = (int)ei[e];
    int dst = (int)ei[(long long)E + e];
    const float4 v = *(const float4*)(x + (size_t)src * DIN + lane * 4);
    float* s = sum + (size_t)dst * DIN + lane * 4;
    (void)__hip_atomic_fetch_add(&s[0], v.x, __ATOMIC_RELAXED, __HIP_MEMORY_SCOPE_AGENT);
    (void)__hip_atomic_fetch_add(&s[1], v.y, __ATOMIC_RELAXED, __HIP_MEMORY_SCOPE_AGENT);
    (void)__hip_atomic_fetch_add(&s[2], v.z, __ATOMIC_RELAXED, __HIP_MEMORY_SCOPE_AGENT);
    (void)__hip_atomic_fetch_add(&s[3], v.w, __ATOMIC_RELAXED, __HIP_MEMORY_SCOPE_AGENT);
  }
}

// sum[i][d] /= max(deg[i], 1)   (in place; D = 128 so index>>7 gives node)
__global__ void mean_kernel(float* __restrict__ sum,
                            const float* __restrict__ deg, long long n) {
  long long i = (long long)blockIdx.x * blockDim.x + threadIdx.x;
  if (i < n) {
    float d = deg[i >> 7];
    sum[i] = sum[i] / fmaxf(d, 1.0f);
  }
}

// WT[n*K + k] = W[k*ncols + n]  (run once per launch, tiny)
__global__ void transpose_kernel(const float* __restrict__ W,
                                 float* __restrict__ WT, int K, int ncols) {
  int i = blockIdx.x * blockDim.x + threadIdx.x;
  if (i < K * ncols) {
    int k = i / ncols, n = i % ncols;
    WT[n * K + k] = W[i];
  }
}

// TDM: 1-D copy of `nelem` f32 from global `ga` into LDS byte-offset `ldsoff`.
// D# per CDNA5 ISA 8.3/8.4: group0 {count=1, lds_addr, global_addr[56:0],
// type=2}; group1 {data_size=4B, tensor_dim0=tile_dim0=nelem (1-D tile)}.
// clang-23 toolchain: 6-arg builtin (g0, g1, g2, g3, g4, cpol).
__device__ __forceinline__ void tdm_load_1d(unsigned int ldsoff,
                                            unsigned long long ga,
                                            unsigned int nelem) {
  u32x4 g0;
  g0[0] = 1u;                                                // count = 1
  g0[1] = ldsoff;                                            // lds_addr
  g0[2] = (unsigned int)ga;                                  // global_addr[31:0]
  g0[3] = ((unsigned int)(ga >> 32) & 0x01FFFFFFu)           // global_addr[56:32]
          | (2u << 30);                                      // type = 2 (image)
  i32x8 g1;
  g1[0] = (int)(2u << 16);                                   // data_size = 4B
  g1[1] = (int)((nelem & 0xFFFFu) << 16);                    // tensor_dim0[15:0]
  g1[2] = (int)((nelem >> 16) & 0xFFFFu);                    // tensor_dim0[31:16]
  g1[3] = (int)((nelem & 0xFFFFu) << 16);                    // tile_dim0
  g1[4] = 0;                                                 // tile_dim1/2 = 0 (1-D)
  g1[5] = (int)nelem;                                        // dim0_stride[31:0]
  g1[6] = 0;
  g1[7] = 0;
  i32x4 gz4 = {0, 0, 0, 0};
  i32x8 gz8 = {0, 0, 0, 0, 0, 0, 0, 0};
  __builtin_amdgcn_tensor_load_to_lds(g0, g1, gz4, gz4, gz8, 0);
}

// Fused GEMM:  out = [relu]( Amean @ Wl + Aself @ Wr + bias )
// WT = both weights transposed+contiguous: WT[0..ncols*K) = Wl^T, then Wr^T.
// One wave per 16x16 f32 output tile via V_WMMA_F32_16X16X4_F32; B fragments
// become contiguous v2f in LDS -> single ds_load_b64 w/ immediate offsets.
__global__ void sage_gemm_kernel(const float* __restrict__ Amean,
                                 const float* __restrict__ Aself,
                                 const float* __restrict__ WT,
                                 const float* __restrict__ bias,
                                 float* __restrict__ out,
                                 int ncols, int do_relu) {
  extern __shared__ float lds[];          // [2][ncols][K] transposed weights
  const int K = 128;
  const int total = K * ncols;            // elements per matrix

  if (threadIdx.x < 32) {                 // wave 0 issues the DMA
    tdm_load_1d((unsigned int)(size_t)lds,
                (unsigned long long)(size_t)WT,
                (unsigned int)(2 * total));
    __builtin_amdgcn_s_wait_tensorcnt(0);
  }
  __syncthreads();

  const int lane  = threadIdx.x & 31;
  const int wave  = threadIdx.x >> 5;
  const int m15   = lane & 15;
  const int khalf = lane >> 4;            // 0: K pair {k,k+1}; 1: {k+2,k+3}
  const int row   = blockIdx.x * 16 + m15;   // N = 6250 * 16 exactly
  const int n     = wave * 16 + m15;         // output column

  const float* a0 = Amean + (size_t)row * K + khalf * 2;
  const float* a1 = Aself + (size_t)row * K + khalf * 2;
  const float* bl = lds + n * K + khalf * 2;          // Wl^T row n
  const float* br = bl + total;                       // Wr^T row n

  v8f acc = {};

  #pragma unroll
  for (int k = 0; k < K; k += 4) {        // mean @ Wl
    v2f a = *(const v2f*)(a0 + k);
    v2f b = *(const v2f*)(bl + k);
    acc = __builtin_amdgcn_wmma_f32_16x16x4_f32(
        false, a, false, b, (short)0, acc, false, false);
  }
  #pragma unroll
  for (int k = 0; k < K; k += 4) {        // x @ Wr (fused K-extension)
    v2f a = *(const v2f*)(a1 + k);
    v2f b = *(const v2f*)(br + k);
    acc = __builtin_amdgcn_wmma_f32_16x16x4_f32(
        false, a, false, b, (short)0, acc, false, false);
  }

  const float bv = bias[n];
  const int mbase = blockIdx.x * 16 + khalf * 8;
  #pragma unroll
  for (int i = 0; i < 8; ++i) {
    float v = acc[i] + bv;
    if (do_relu) v = fmaxf(v, 0.0f);
    out[(size_t)(mbase + i) * ncols + n] = v;
  }
}

// ---------------------------------------------------------------- launcher
extern "C" void kernel_launch(void* const* d_in, const int* in_sizes, int n_in,
                              void* d_out, int out_size, void* d_ws, size_t ws_size,
                              hipStream_t stream) {
  const float*     x   = (const float*)d_in[0];
  const long long* ei  = (const long long*)d_in[1];   // int64 edge_index [2,E]
  const float*     W1l = (const float*)d_in[2];
  const float*     W1r = (const float*)d_in[3];
  const float*     b1  = (const float*)d_in[4];
  const float*     W2l = (const float*)d_in[5];
  const float*     W2r = (const float*)d_in[6];
  const float*     b2  = (const float*)d_in[7];
  float*           out = (float*)d_out;

  // workspace: deg[N] | sum[N*128] | h[N*128] | W1^T pair | W2^T pair
  float* deg = (float*)d_ws;
  float* sum = deg + NNODES;
  float* h   = sum + (size_t)NNODES * DHID;
  float* w1t = h   + (size_t)NNODES * DHID;   // 2 * 128*128
  float* w2t = w1t + 2 * DIN * DHID;          // 2 * 128*64

  const long long nf = (long long)NNODES * DHID;
  const int zb = 256;

  // ---- one-time per launch: transpose weights (deterministic, tiny) ----
  transpose_kernel<<<(DIN * DHID + zb - 1) / zb, zb, 0, stream>>>(W1l, w1t, DIN, DHID);
  transpose_kernel<<<(DIN * DHID + zb - 1) / zb, zb, 0, stream>>>(W1r, w1t + DIN * DHID, DIN, DHID);
  transpose_kernel<<<(DHID * DOUT + zb - 1) / zb, zb, 0, stream>>>(W2l, w2t, DHID, DOUT);
  transpose_kernel<<<(DHID * DOUT + zb - 1) / zb, zb, 0, stream>>>(W2r, w2t + DHID * DOUT, DHID, DOUT);

  // ---- layer 1 ----
  zero_kernel<<<(NNODES + zb - 1) / zb, zb, 0, stream>>>(deg, NNODES);
  zero_kernel<<<(int)((nf + zb - 1) / zb), zb, 0, stream>>>(sum, nf);
  degree_kernel<<<(NEDGES + zb - 1) / zb, zb, 0, stream>>>(ei, deg, NEDGES);
  scatter_kernel<<<(NEDGES + 7) / 8, 256, 0, stream>>>(x, ei, sum, NEDGES);
  mean_kernel<<<(int)((nf + zb - 1) / zb), zb, 0, stream>>>(sum, deg, nf);
  sage_gemm_kernel<<<NNODES / 16, 256, 2 * DIN * DHID * sizeof(float), stream>>>(
      sum, x, w1t, b1, h, DHID, 1);

  // ---- layer 2 ----
  zero_kernel<<<(int)((nf + zb - 1) / zb), zb, 0, stream>>>(sum, nf);
  scatter_kernel<<<(NEDGES + 7) / 8, 256, 0, stream>>>(h, ei, sum, NEDGES);
  mean_kernel<<<(int)((nf + zb - 1) / zb), zb, 0, stream>>>(sum, deg, nf);
  sage_gemm_kernel<<<NNODES / 16, 128, 2 * DHID * DOUT * sizeof(float), stream>>>(
      sum, h, w2t, b2, out, DOUT, 0);
}